// GlobalPointer_72584947302407
// MI455X (gfx1250) — compile-verified
//
#include <hip/hip_runtime.h>
#include <hip/hip_bf16.h>
#include <math.h>

typedef __attribute__((ext_vector_type(2))) float v2f;
typedef __attribute__((ext_vector_type(8))) float v8f;

#define NSEQ 512
#define NCH  108
#define DHEAD 64
#define HIN  768
#define PLANE (NSEQ*NSEQ)   /* 262144 = 1<<18 */

// ---------------------------------------------------------------------------
// K0: zero the fused-reduction accumulators (graph-replay deterministic)
// ---------------------------------------------------------------------------
__global__ __launch_bounds__(256) void k0_init(
    float* __restrict__ chanSum, float* __restrict__ chanMax)
{
    const int tid = threadIdx.x;
    if (tid < 2*NCH) { chanSum[tid] = 0.f; chanMax[tid] = -__builtin_inff(); }
}

// ---------------------------------------------------------------------------
// K1: seq = inputs @ w_dense + b_dense, then RoPE -> qw, kw
// ---------------------------------------------------------------------------
__global__ __launch_bounds__(128) void k1_seq_rope(
    const float* __restrict__ in, const float* __restrict__ wd,
    const float* __restrict__ bd, float* __restrict__ qw, float* __restrict__ kw)
{
    __shared__ float row[HIN];
    __shared__ float sq[128];
    const int tid = threadIdx.x;
    const int bi  = blockIdx.x / NSEQ;
    const int pos = blockIdx.x % NSEQ;
    const float* rp = in + (size_t)(bi*NSEQ + pos) * HIN;
    for (int i = tid; i < HIN; i += 128) row[i] = rp[i];
    __syncthreads();
    float acc = 0.f;
    #pragma unroll 4
    for (int k = 0; k < HIN; ++k) acc += row[k] * wd[k*128 + tid];
    sq[tid] = acc + bd[tid];
    __syncthreads();
    const int d = tid & 63;
    const int pair = d >> 1;
    const float ang = (float)pos * expf(-(float)(2*pair) * (9.210340371976184f/64.f));
    const float cs = cosf(ang), sn = sinf(ang);
    const float x  = sq[tid];
    const float xp = sq[tid ^ 1];
    const float r = (d & 1) ? fmaf(x, cs, xp*sn) : fmaf(x, cs, -xp*sn);
    float* dst = (tid < 64) ? qw : kw;
    dst[(size_t)(bi*NSEQ + pos)*DHEAD + d] = r;
}

// ---------------------------------------------------------------------------
// K1b: ev = event_inputs @ w_event + b_event   (108 x 64)
// ---------------------------------------------------------------------------
__global__ __launch_bounds__(64) void k1b_ev(
    const float* __restrict__ ein, const float* __restrict__ we,
    const float* __restrict__ be, float* __restrict__ ev)
{
    __shared__ float row[HIN];
    const int tid = threadIdx.x;
    const int c = blockIdx.x;
    const float* rp = ein + (size_t)c * HIN;
    for (int i = tid; i < HIN; i += 64) row[i] = rp[i];
    __syncthreads();
    float acc = 0.f;
    #pragma unroll 4
    for (int k = 0; k < HIN; ++k) acc += row[k] * we[k*DHEAD + tid];
    ev[c*DHEAD + tid] = acc + be[tid];
}

// ---------------------------------------------------------------------------
// K2: logits[b,c,m,n] = sum_d (qw[b,m,d]*ev[c,d]) * kw[b,n,d]
// fp32 WMMA 16x16x4, BM=64 BN=128 per block, 8 waves x (1M x 4N) subtiles.
// Fused epilogue: block-level sum/max -> atomicAdd / atomicMax per (b,c).
// ---------------------------------------------------------------------------
__global__ __launch_bounds__(256) void k2_gemm_wmma(
    const float* __restrict__ qw, const float* __restrict__ kw,
    const float* __restrict__ ev, float* __restrict__ out,
    float* __restrict__ chanSum, float* __restrict__ chanMax)
{
    __shared__ float qs[64*65];
    __shared__ float ks[128*65];
    __shared__ float es[64];
    const int tid = threadIdx.x;
    const int z = blockIdx.z;            // b*108 + c
    const int bIdx = z / NCH;
    const int cIdx = z % NCH;
    const int m0 = blockIdx.x * 64;
    const int n0 = blockIdx.y * 128;

    const float* qbase = qw + (size_t)(bIdx*NSEQ + m0) * DHEAD;
    for (int i = tid; i < 1024; i += 256) {          // 64 rows x 16 float4
        const int r = i >> 4, c4 = (i & 15) << 2;
        const float4 v = *(const float4*)(qbase + (size_t)r*DHEAD + c4);
        float* dst = &qs[r*65 + c4];
        dst[0]=v.x; dst[1]=v.y; dst[2]=v.z; dst[3]=v.w;
    }
    const float* kbase = kw + (size_t)(bIdx*NSEQ + n0) * DHEAD;
    for (int i = tid; i < 2048; i += 256) {          // 128 rows x 16 float4
        const int r = i >> 4, c4 = (i & 15) << 2;
        const float4 v = *(const float4*)(kbase + (size_t)r*DHEAD + c4);
        float* dst = &ks[r*65 + c4];
        dst[0]=v.x; dst[1]=v.y; dst[2]=v.z; dst[3]=v.w;
    }
    if (tid < 16) {
        const float4 v = *(const float4*)(ev + cIdx*DHEAD + tid*4);
        es[tid*4+0]=v.x; es[tid*4+1]=v.y; es[tid*4+2]=v.z; es[tid*4+3]=v.w;
    }
    __syncthreads();

    const int wave  = tid >> 5;
    const int lane  = tid & 31;
    const int lm    = lane & 15;
    const int khalf = (lane >> 4) << 1;   // lanes0-15: K={k0,k0+1}, lanes16-31: K={k0+2,k0+3}
    const int mw = (wave & 3) * 16;       // local M subtile base
    const int nw = (wave >> 2) * 64;      // local N strip base

    v8f acc[4] = {};
    #pragma unroll
    for (int k0 = 0; k0 < DHEAD; k0 += 4) {
        const int kk = k0 + khalf;
        v2f a;
        a.x = qs[(mw + lm)*65 + kk]     * es[kk];
        a.y = qs[(mw + lm)*65 + kk + 1] * es[kk + 1];
        #pragma unroll
        for (int s = 0; s < 4; ++s) {
            const int nl = nw + s*16 + lm;
            v2f bf;
            bf.x = ks[nl*65 + kk];
            bf.y = ks[nl*65 + kk + 1];
            acc[s] = __builtin_amdgcn_wmma_f32_16x16x4_f32(
                false, a, false, bf, (short)0, acc[s], false, false);
        }
    }

    // C/D layout: VGPR r -> row = base + r + (lane>=16 ? 8 : 0), col = lane&15
    const size_t plane = (size_t)z * (size_t)PLANE;
    const int rowBase = m0 + mw + ((lane >> 4) << 3);
    const int colBase = n0 + nw + lm;
    float ls = 0.f, lmx = -__builtin_inff();
    #pragma unroll
    for (int s = 0; s < 4; ++s) {
        const int col = colBase + s*16;
        #pragma unroll
        for (int r = 0; r < 8; ++r) {
            const float v = acc[s][r];
            out[plane + (size_t)(rowBase + r)*NSEQ + col] = v;
            ls += v;
            lmx = fmaxf(lmx, v);
        }
    }

    // fused channel-stats reduction (reuse qs after all LDS readers are done)
    __syncthreads();
    qs[tid] = ls;
    qs[256 + tid] = lmx;
    __syncthreads();
    for (int o = 128; o > 0; o >>= 1) {
        if (tid < o) {
            qs[tid] += qs[tid + o];
            qs[256 + tid] = fmaxf(qs[256 + tid], qs[256 + tid + o]);
        }
        __syncthreads();
    }
    if (tid == 0) {
        atomicAdd(&chanSum[z], qs[0]);
        atomicMax(&chanMax[z], qs[256]);
    }
}

// ---------------------------------------------------------------------------
// K3b: channel-attention MLP: scale = sigmoid(mlp(avg)+mlp(mx)), 1 block
// ---------------------------------------------------------------------------
__global__ __launch_bounds__(256) void k3b_scale(
    const float* __restrict__ chanSum, const float* __restrict__ mx,
    const float* __restrict__ w1, const float* __restrict__ b1,
    const float* __restrict__ w2, const float* __restrict__ b2,
    float* __restrict__ scale)
{
    __shared__ float hb[2][2][6];
    const int tid = threadIdx.x;
    if (tid < 24) {
        const int b = tid / 12, rem = tid % 12, which = rem / 6, j = rem % 6;
        float a = b1[j];
        if (which) {
            const float* v = mx + b*NCH;
            for (int c = 0; c < NCH; ++c) a += v[c] * w1[c*6 + j];
        } else {
            const float* v = chanSum + b*NCH;
            for (int c = 0; c < NCH; ++c) a += (v[c] * (1.f/PLANE)) * w1[c*6 + j];
        }
        hb[b][which][j] = fmaxf(a, 0.f);
    }
    __syncthreads();
    if (tid < 2*NCH) {
        const int b = tid / NCH, c = tid % NCH;
        float a = 2.f * b2[c];
        for (int j = 0; j < 6; ++j) a += (hb[b][0][j] + hb[b][1][j]) * w2[j*NCH + c];
        scale[tid] = 1.f / (1.f + expf(-a));
    }
}

// ---------------------------------------------------------------------------
// K3c: per-pixel max & mean over channels of scaled logits (float4)
// ---------------------------------------------------------------------------
__global__ __launch_bounds__(256) void k3c_comp(
    const float* __restrict__ out, const float* __restrict__ scale,
    float* __restrict__ compMax, float* __restrict__ compMean)
{
    const int q = blockIdx.x*256 + threadIdx.x;   // 0..131071 quads
    const int p = q << 2;
    const int b = p >> 18, hw = p & (PLANE-1);
    const float* base = out + (((size_t)b*NCH) << 18) + hw;
    const float* sc = scale + b*NCH;
    float4 m = make_float4(-__builtin_inff(), -__builtin_inff(),
                           -__builtin_inff(), -__builtin_inff());
    float4 s = make_float4(0.f, 0.f, 0.f, 0.f);
    for (int c = 0; c < NCH; ++c) {
        const float4 v = *(const float4*)(base + ((size_t)c << 18));
        const float sv = sc[c];
        m.x = fmaxf(m.x, v.x*sv); m.y = fmaxf(m.y, v.y*sv);
        m.z = fmaxf(m.z, v.z*sv); m.w = fmaxf(m.w, v.w*sv);
        s.x += v.x*sv; s.y += v.y*sv; s.z += v.z*sv; s.w += v.w*sv;
    }
    *(float4*)(compMax + p) = m;
    const float inv = 1.f/NCH;
    s.x *= inv; s.y *= inv; s.z *= inv; s.w *= inv;
    *(float4*)(compMean + p) = s;
}

// ---------------------------------------------------------------------------
// K4: 7x7 spatial conv (2 in-ch: [max, mean]) -> sigmoid gate per pixel
// ---------------------------------------------------------------------------
__global__ __launch_bounds__(256) void k4_gate(
    const float* __restrict__ compMax, const float* __restrict__ compMean,
    const float* __restrict__ cw, float* __restrict__ gate)
{
    __shared__ float w[98];
    const int tid = threadIdx.x;
    if (tid < 98) w[tid] = cw[tid];
    __syncthreads();
    const int p = blockIdx.x*256 + tid;
    const int b = p >> 18, hw = p & (PLANE-1);
    const int h = hw >> 9, x = hw & 511;
    float a = 0.f;
    for (int ic = 0; ic < 2; ++ic) {
        const float* src = (ic == 0 ? compMax : compMean) + ((size_t)b << 18);
        for (int kh = 0; kh < 7; ++kh) {
            const int hh = h + kh - 3;
            if (hh < 0 || hh >= NSEQ) continue;
            for (int kw7 = 0; kw7 < 7; ++kw7) {
                const int cc = x + kw7 - 3;
                if (cc < 0 || cc >= NSEQ) continue;
                a += src[hh*NSEQ + cc] * w[(ic*7 + kh)*7 + kw7];
            }
        }
    }
    a /= sqrtf(1.0f + 1e-5f);
    gate[p] = 1.f / (1.f + expf(-a));
}

// ---------------------------------------------------------------------------
// K5: out = (x*scale*gate + x), mask -> -inf, tril penalty, /8  (float4)
// ---------------------------------------------------------------------------
__global__ __launch_bounds__(256) void k5_final(
    float* __restrict__ out, const float* __restrict__ scale,
    const float* __restrict__ gate, const int* __restrict__ mask)
{
    const size_t e = ((size_t)blockIdx.x*256 + threadIdx.x) << 2;
    const int n  = (int)(e & 511);          // n..n+3, same row
    const int m  = (int)((e >> 9) & 511);
    const int bc = (int)(e >> 18);
    const int b  = bc / NCH;
    const float sc = scale[bc];
    const int p = (b << 18) | (m << 9) | n;
    const float4 x = *(const float4*)(out + e);
    const float4 g = *(const float4*)(gate + p);
    float4 v;
    v.x = fmaf(x.x*sc, g.x, x.x);
    v.y = fmaf(x.y*sc, g.y, x.y);
    v.z = fmaf(x.z*sc, g.z, x.z);
    v.w = fmaf(x.w*sc, g.w, x.w);
    const int rowOk = mask[b*NSEQ + m];
    const float ninf = -__builtin_inff();
    if (!rowOk || !mask[b*NSEQ + n + 0]) v.x = ninf;
    if (!rowOk || !mask[b*NSEQ + n + 1]) v.y = ninf;
    if (!rowOk || !mask[b*NSEQ + n + 2]) v.z = ninf;
    if (!rowOk || !mask[b*NSEQ + n + 3]) v.w = ninf;
    if (m > n + 0) v.x -= 1e12f;
    if (m > n + 1) v.y -= 1e12f;
    if (m > n + 2) v.z -= 1e12f;
    if (m > n + 3) v.w -= 1e12f;
    v.x *= 0.125f; v.y *= 0.125f; v.z *= 0.125f; v.w *= 0.125f;
    *(float4*)(out + e) = v;
}

// ---------------------------------------------------------------------------
extern "C" void kernel_launch(void* const* d_in, const int* in_sizes, int n_in,
                              void* d_out, int out_size, void* d_ws, size_t ws_size,
                              hipStream_t stream)
{
    const float* inputs       = (const float*)d_in[0];
    const float* event_inputs = (const float*)d_in[1];
    const int*   mask         = (const int*)  d_in[2];
    const float* w_dense      = (const float*)d_in[3];
    const float* b_dense      = (const float*)d_in[4];
    const float* w_event      = (const float*)d_in[5];
    const float* b_event      = (const float*)d_in[6];
    const float* mlp_w1       = (const float*)d_in[7];
    const float* mlp_b1       = (const float*)d_in[8];
    const float* mlp_w2       = (const float*)d_in[9];
    const float* mlp_b2       = (const float*)d_in[10];
    const float* conv_w       = (const float*)d_in[11];
    float* out = (float*)d_out;
    float* ws  = (float*)d_ws;

    float* qw       = ws;            // 65536
    float* kw       = ws + 65536;    // 65536
    float* ev       = ws + 131072;   // 6912
    float* chanSum  = ws + 139264;   // 216
    float* chanMax  = ws + 139520;   // 216
    float* scale    = ws + 139776;   // 216
    float* compMax  = ws + 140288;   // 524288
    float* compMean = ws + 664576;   // 524288
    float* gate     = ws + 1188864;  // 524288  (end ~6.9 MB)

    k0_init      <<<dim3(1),         dim3(256), 0, stream>>>(chanSum, chanMax);
    k1_seq_rope  <<<dim3(1024),      dim3(128), 0, stream>>>(inputs, w_dense, b_dense, qw, kw);
    k1b_ev       <<<dim3(108),       dim3(64),  0, stream>>>(event_inputs, w_event, b_event, ev);
    k2_gemm_wmma <<<dim3(8, 4, 216), dim3(256), 0, stream>>>(qw, kw, ev, out, chanSum, chanMax);
    k3b_scale    <<<dim3(1),         dim3(256), 0, stream>>>(chanSum, chanMax, mlp_w1, mlp_b1, mlp_w2, mlp_b2, scale);
    k3c_comp     <<<dim3(512),       dim3(256), 0, stream>>>(out, scale, compMax, compMean);
    k4_gate      <<<dim3(2048),      dim3(256), 0, stream>>>(compMax, compMean, conv_w, gate);
    k5_final     <<<dim3(55296),     dim3(256), 0, stream>>>(out, scale, gate, mask);
}